// LowRankProjection_20349555048714
// MI455X (gfx1250) — compile-verified
//
#include <hip/hip_runtime.h>

// -------- problem constants (from reference) --------
#define B_DIM   4096
#define N_PRE   16384
#define N_POST  16384
#define R_DIM   32

typedef __attribute__((ext_vector_type(2))) float v2f;
typedef __attribute__((ext_vector_type(8))) float v8f;

// ============================================================
// Phase 1: z[B, R] = spikes[B, N_PRE] @ V[N_PRE, R]
//   grid.x = B/16 row tiles, 256 threads = 8 waves.
//   Each wave owns a K slice of 2048, accumulates a 16x32 tile
//   with V_WMMA_F32_16X16X4_F32, then 8 partials reduced via LDS.
// ============================================================
__global__ __launch_bounds__(256)
void lr_phase1(const float* __restrict__ spikes,
               const float* __restrict__ V,
               float* __restrict__ z)
{
    const int lane = threadIdx.x & 31;
    const int wave = threadIdx.x >> 5;
    const int m    = lane & 15;          // row within tile (A) / col within tile (B)
    const int kh   = (lane >> 4) << 1;   // 0 for lanes 0-15, 2 for lanes 16-31

    const int row0 = blockIdx.x * 16;
    const int KW   = N_PRE / 8;          // 2048 K per wave
    const int kbeg = wave * KW;

    const float* __restrict__ arow = spikes + (size_t)(row0 + m) * N_PRE;

    v8f c0 = {};   // cols 0..15
    v8f c1 = {};   // cols 16..31

    for (int k0 = kbeg; k0 < kbeg + KW; k0 += 32) {
        // stream-prefetch the spikes row one tile group ahead
        __builtin_prefetch(arow + k0 + 512, 0, 0);
#pragma unroll
        for (int kk = 0; kk < 32; kk += 4) {
            const int k = k0 + kk;
            // A fragment: 16x4 f32 (lane m holds spikes[row0+m][k+kh .. k+kh+1])
            v2f a = *(const v2f*)(arow + k + kh);
            // B fragment: 4x16 f32 rows of V (b.x = V[k+kh][n], b.y = V[k+kh+1][n])
            const float* __restrict__ vr = V + (size_t)(k + kh) * R_DIM;
            v2f b0, b1;
            b0.x = vr[m];             b0.y = vr[R_DIM + m];
            b1.x = vr[16 + m];        b1.y = vr[R_DIM + 16 + m];
            c0 = __builtin_amdgcn_wmma_f32_16x16x4_f32(false, a, false, b0,
                                                       (short)0, c0, false, false);
            c1 = __builtin_amdgcn_wmma_f32_16x16x4_f32(false, a, false, b1,
                                                       (short)0, c1, false, false);
        }
    }

    // ---- reduce the 8 wave partials through LDS ----
    __shared__ float red[8][16 * 32];
    const int mrow = (lane >> 4) << 3;   // C layout: M = v + 8*(lane/16)
#pragma unroll
    for (int v = 0; v < 8; ++v) {
        red[wave][(v + mrow) * 32 + m]      = c0[v];
        red[wave][(v + mrow) * 32 + 16 + m] = c1[v];
    }
    __syncthreads();

#pragma unroll
    for (int e = threadIdx.x; e < 16 * 32; e += 256) {
        float s = 0.f;
#pragma unroll
        for (int w = 0; w < 8; ++w) s += red[w][e];
        z[(size_t)(row0 + (e >> 5)) * R_DIM + (e & 31)] = s;
    }
}

// ============================================================
// Phase 2: y[B, N_POST] = z[B, R] @ U^T   (U is [N_POST, R])
//   grid = (N_POST/64, B/128); 256 threads = 8 waves.
//   Each wave: one 16-row tile x 64 cols = 4 N-tiles, K=32 -> 32 WMMAs.
//   A fragments (z) preloaded once, reused for all 4 N-tiles.
// ============================================================
__global__ __launch_bounds__(256)
void lr_phase2(const float* __restrict__ z,
               const float* __restrict__ U,
               float* __restrict__ y)
{
    const int lane = threadIdx.x & 31;
    const int wave = threadIdx.x >> 5;
    const int m    = lane & 15;
    const int kh   = (lane >> 4) << 1;

    const int row0 = blockIdx.y * 128 + wave * 16;
    const int col0 = blockIdx.x * 64;

    // Preload all 8 A fragments of the z tile (K = 0..31)
    const float* __restrict__ zrow = z + (size_t)(row0 + m) * R_DIM + kh;
    v2f a[8];
#pragma unroll
    for (int kk = 0; kk < 8; ++kk) a[kk] = *(const v2f*)(zrow + kk * 4);

    v8f c[4] = {v8f{}, v8f{}, v8f{}, v8f{}};
#pragma unroll
    for (int t = 0; t < 4; ++t) {
        // B fragment: B[k][n] = U[col0 + t*16 + n][k]; lane n reads a contiguous
        // 8-byte pair from U's row -> global_load_b64
        const float* __restrict__ ur =
            U + (size_t)(col0 + t * 16 + m) * R_DIM + kh;
#pragma unroll
        for (int kk = 0; kk < 8; ++kk) {
            v2f b = *(const v2f*)(ur + kk * 4);
            c[t] = __builtin_amdgcn_wmma_f32_16x16x4_f32(false, a[kk], false, b,
                                                         (short)0, c[t], false, false);
        }
    }

    // ---- store 16x64 tile; each (t,v) store covers 2 x 64 contiguous bytes ----
    const int mrow = (lane >> 4) << 3;
#pragma unroll
    for (int t = 0; t < 4; ++t) {
#pragma unroll
        for (int v = 0; v < 8; ++v) {
            y[(size_t)(row0 + v + mrow) * N_POST + col0 + t * 16 + m] = c[t][v];
        }
    }
}

// ============================================================
extern "C" void kernel_launch(void* const* d_in, const int* in_sizes, int n_in,
                              void* d_out, int out_size, void* d_ws, size_t ws_size,
                              hipStream_t stream)
{
    const float* spikes = (const float*)d_in[0];   // [B, N_PRE]
    const float* U      = (const float*)d_in[1];   // [N_POST, R]
    const float* V      = (const float*)d_in[2];   // [N_PRE, R]
    // d_in[3..5] = CSR mask buffers: dead in the reference, ignored.

    float* y = (float*)d_out;                      // [B, N_POST]
    float* z = (float*)d_ws;                       // [B, R] scratch (512 KB)

    lr_phase1<<<dim3(B_DIM / 16), dim3(256), 0, stream>>>(spikes, V, z);
    lr_phase2<<<dim3(N_POST / 64, B_DIM / 128), dim3(256), 0, stream>>>(z, U, y);
}